// TopkSAGEBlock_29334626631947
// MI455X (gfx1250) — compile-verified
//
#include <hip/hip_runtime.h>
#include <math.h>

#define Nn   65536
#define Ee   524288
#define Bb   64
#define PERg 1024
#define HF   128
#define KK   820               // ceil(0.8 * 1024)
#define BK   (Bb * KK)         // 52480
#define EPSv 1e-5f
#define GROWS 64               // rows of h per GEMM block

typedef float v2f __attribute__((ext_vector_type(2)));
typedef float v8f __attribute__((ext_vector_type(8)));

// ---------------------------------------------------------------------------
// K0: zero accumulators, node_map = -1 (must run every call; ws not re-poisoned)
// ---------------------------------------------------------------------------
__global__ __launch_bounds__(256) void k0_init(float* __restrict__ hbuf,
                                               float* __restrict__ cnt,
                                               float* __restrict__ colstats,
                                               int* __restrict__ node_map) {
  int i = blockIdx.x * 256 + threadIdx.x;
  if (i < Nn * HF) hbuf[i] = 0.0f;
  if (i < Nn) { cnt[i] = 0.0f; node_map[i] = -1; }
  if (i < 2 * HF) colstats[i] = 0.0f;
}

// ---------------------------------------------------------------------------
// K1: edge scatter  agg[dst] += x[src];  cnt[dst] += 1
// block = 256 threads = 2 edges x 128 features
// ---------------------------------------------------------------------------
__global__ __launch_bounds__(256) void k1_scatter(const float* __restrict__ x,
                                                  const int* __restrict__ eidx,
                                                  float* __restrict__ agg,
                                                  float* __restrict__ cnt) {
  int e = blockIdx.x * 2 + (threadIdx.x >> 7);
  int f = threadIdx.x & (HF - 1);
  int s = eidx[e];
  int d = eidx[Ee + e];
  atomicAdd(&agg[d * HF + f], x[s * HF + f]);
  if (f == 0) atomicAdd(&cnt[d], 1.0f);
}

// ---------------------------------------------------------------------------
// K2: h = (agg / max(cnt,1)) @ Wl^T + x @ Wr^T + bl   via V_WMMA_F32_16X16X4_F32
// Combined K = 256 ( [agg_norm | x] @ [Wl^T ; Wr^T] ). agg buffer is reused
// in-place for h: each block stages its 64 rows into LDS before writing.
// ---------------------------------------------------------------------------
__global__ __launch_bounds__(256) void k2_gemm(const float* __restrict__ x,
                                               const float* __restrict__ Wl,
                                               const float* __restrict__ bl,
                                               const float* __restrict__ Wr,
                                               const float* __restrict__ cnt,
                                               float* __restrict__ hbuf) {
  __shared__ float At[GROWS][258];          // stride 258: avoids 64-bank conflicts
  const int row0 = blockIdx.x * GROWS;
  const int tid = threadIdx.x;

  // Stage A tile: cols 0..127 = normalized agg, cols 128..255 = x
  for (int i = tid; i < GROWS * 256; i += 256) {
    int r = i >> 8, c = i & 255;
    int g = row0 + r;
    float v;
    if (c < HF) v = hbuf[g * HF + c] / fmaxf(cnt[g], 1.0f);
    else        v = x[g * HF + (c - HF)];
    At[r][c] = v;
  }
  __syncthreads();

  const int lane  = tid & 31;
  const int wave  = tid >> 5;              // 8 waves -> 8 column tiles of 16
  const int m     = lane & 15;
  const int kkoff = (lane >> 4) << 1;      // lanes 0-15: K{0,1}; 16-31: K{2,3}
  const int ncol  = wave * 16 + m;         // output column / weight row
  const float* wl = Wl + ncol * HF;
  const float* wr = Wr + ncol * HF;

  v8f acc0 = {}, acc1 = {}, acc2 = {}, acc3 = {};
  for (int k0 = 0; k0 < 256; k0 += 4) {
    int kb = k0 + kkoff;                   // boundary 128 is a multiple of 4 -> uniform select
    const float* wsrc = (kb < HF) ? (wl + kb) : (wr + (kb - HF));
    v2f b; b.x = wsrc[0]; b.y = wsrc[1];   // B[k][n] = Wcomb[n][k]
    v2f a0, a1, a2, a3;
    a0.x = At[m     ][kb]; a0.y = At[m     ][kb + 1];
    a1.x = At[16 + m][kb]; a1.y = At[16 + m][kb + 1];
    a2.x = At[32 + m][kb]; a2.y = At[32 + m][kb + 1];
    a3.x = At[48 + m][kb]; a3.y = At[48 + m][kb + 1];
    acc0 = __builtin_amdgcn_wmma_f32_16x16x4_f32(false, a0, false, b, (short)0, acc0, false, false);
    acc1 = __builtin_amdgcn_wmma_f32_16x16x4_f32(false, a1, false, b, (short)0, acc1, false, false);
    acc2 = __builtin_amdgcn_wmma_f32_16x16x4_f32(false, a2, false, b, (short)0, acc2, false, false);
    acc3 = __builtin_amdgcn_wmma_f32_16x16x4_f32(false, a3, false, b, (short)0, acc3, false, false);
  }

  const float bias = bl[ncol];
  const int rb = (lane >> 4) * 8;          // D layout: lanes 16-31 hold M+8
#pragma unroll
  for (int v = 0; v < 8; ++v) {
    hbuf[(row0 +      rb + v) * HF + ncol] = acc0[v] + bias;
    hbuf[(row0 + 16 + rb + v) * HF + ncol] = acc1[v] + bias;
    hbuf[(row0 + 32 + rb + v) * HF + ncol] = acc2[v] + bias;
    hbuf[(row0 + 48 + rb + v) * HF + ncol] = acc3[v] + bias;
  }
}

// ---------------------------------------------------------------------------
// K3: per-column sum / sum-of-squares partials -> atomics into colstats
// ---------------------------------------------------------------------------
__global__ __launch_bounds__(256) void k3_stats(const float* __restrict__ hbuf,
                                                float* __restrict__ colsum,
                                                float* __restrict__ colsumsq) {
  int c = threadIdx.x & (HF - 1);
  int half = threadIdx.x >> 7;
  int base = blockIdx.x * 128;
  float s = 0.0f, s2 = 0.0f;
  for (int i = half; i < 128; i += 2) {
    float v = hbuf[(base + i) * HF + c];
    s += v; s2 += v * v;
  }
  __shared__ float a1[256], a2[256];
  a1[threadIdx.x] = s; a2[threadIdx.x] = s2;
  __syncthreads();
  if (half == 0) {
    s  = a1[threadIdx.x] + a1[threadIdx.x + 128];
    s2 = a2[threadIdx.x] + a2[threadIdx.x + 128];
    atomicAdd(&colsum[c], s);
    atomicAdd(&colsumsq[c], s2);
  }
}

// ---------------------------------------------------------------------------
// K4: BN (training stats, biased var) + ReLU in place, and score[row]
// ---------------------------------------------------------------------------
__global__ __launch_bounds__(128) void k4_bn(const float* __restrict__ colsum,
                                             const float* __restrict__ colsumsq,
                                             const float* __restrict__ gamma,
                                             const float* __restrict__ beta,
                                             const float* __restrict__ attn,
                                             float* __restrict__ hbuf,
                                             float* __restrict__ score) {
  int row = blockIdx.x;
  int c = threadIdx.x;
  float mean = colsum[c] * (1.0f / Nn);
  float var  = colsumsq[c] * (1.0f / Nn) - mean * mean;
  float rstd = rsqrtf(var + EPSv);
  float h = hbuf[row * HF + c];
  h = (h - mean) * rstd * gamma[c] + beta[c];
  h = fmaxf(h, 0.0f);
  hbuf[row * HF + c] = h;
  float a = attn[c];
  __shared__ float r1[HF], r2[HF];
  r1[c] = h * a; r2[c] = a * a;
  __syncthreads();
  for (int off = HF / 2; off > 0; off >>= 1) {
    if (c < off) { r1[c] += r1[c + off]; r2[c] += r2[c + off]; }
    __syncthreads();
  }
  if (c == 0) score[row] = r1[0] * rsqrtf(r2[0]);
}

// ---------------------------------------------------------------------------
// K5: top-K per graph via rank counting (matches lax.top_k tie-breaking)
// ---------------------------------------------------------------------------
__global__ __launch_bounds__(1024) void k5_topk(const float* __restrict__ score,
                                                int* __restrict__ node_map,
                                                int* __restrict__ perm,
                                                float* __restrict__ tscore,
                                                float* __restrict__ out_batch) {
  int g = blockIdx.x;
  int t = threadIdx.x;
  __shared__ float s[PERg];
  float my = score[g * PERg + t];
  s[t] = my;
  __syncthreads();
  int rank = 0;
  for (int j = 0; j < PERg; ++j) {
    float sj = s[j];
    rank += ((sj > my) || (sj == my && j < t)) ? 1 : 0;
  }
  if (rank < KK) {
    int gid = g * PERg + t;
    int pos = g * KK + rank;
    perm[pos] = gid;
    node_map[gid] = pos;
    tscore[pos] = tanhf(my);
    out_batch[pos] = (float)g;
  }
}

// ---------------------------------------------------------------------------
// K6: hp = h[perm] * tanh(score[perm]); per-graph add/max pool -> flat_1
// ---------------------------------------------------------------------------
__global__ __launch_bounds__(1024) void k6_gather(const float* __restrict__ hbuf,
                                                  const int* __restrict__ perm,
                                                  const float* __restrict__ tscore,
                                                  float* __restrict__ out_hp,
                                                  float* __restrict__ out_flat) {
  int g  = blockIdx.x;
  int c  = threadIdx.x & (HF - 1);
  int rs = threadIdx.x >> 7;               // 0..7 row groups
  float sum = 0.0f, mx = -3.402823466e38f;
  for (int r = rs; r < KK; r += 8) {
    int pos = g * KK + r;
    int n = perm[pos];
    float v = hbuf[n * HF + c] * tscore[pos];
    out_hp[pos * HF + c] = v;
    sum += v;
    mx = fmaxf(mx, v);
  }
  __shared__ float ssum[8][HF];
  __shared__ float smax[8][HF];
  ssum[rs][c] = sum; smax[rs][c] = mx;
  __syncthreads();
  if (rs == 0) {
#pragma unroll
    for (int i = 1; i < 8; ++i) { sum += ssum[i][c]; mx = fmaxf(mx, smax[i][c]); }
    out_flat[g * 2 * HF + c]      = sum;
    out_flat[g * 2 * HF + HF + c] = mx;
  }
}

// ---------------------------------------------------------------------------
// K7: edge remap with -1 padding
// ---------------------------------------------------------------------------
__global__ __launch_bounds__(256) void k7_edges(const int* __restrict__ eidx,
                                                const int* __restrict__ node_map,
                                                float* __restrict__ out_edge) {
  int e = blockIdx.x * 256 + threadIdx.x;
  if (e >= Ee) return;
  int ns = node_map[eidx[e]];
  int nd = node_map[eidx[Ee + e]];
  bool keep = (ns >= 0) && (nd >= 0);
  out_edge[e]      = keep ? (float)ns : -1.0f;
  out_edge[Ee + e] = keep ? (float)nd : -1.0f;
}

// ---------------------------------------------------------------------------
extern "C" void kernel_launch(void* const* d_in, const int* in_sizes, int n_in,
                              void* d_out, int out_size, void* d_ws, size_t ws_size,
                              hipStream_t stream) {
  const float* x     = (const float*)d_in[0];
  const int*   eidx  = (const int*)d_in[1];   // [2, E] flat: src then dst
  // d_in[2] = batch (unused: equal-size graphs, PER = N/B)
  const float* Wl    = (const float*)d_in[3];
  const float* bl    = (const float*)d_in[4];
  const float* Wr    = (const float*)d_in[5];
  const float* gamma = (const float*)d_in[6];
  const float* beta  = (const float*)d_in[7];
  const float* attn  = (const float*)d_in[8];

  float* out       = (float*)d_out;
  float* out_hp    = out;                         // [BK, 128]
  float* out_flat  = out_hp + (size_t)BK * HF;    // [64, 256]
  float* out_edge  = out_flat + Bb * 2 * HF;      // [2, E]
  float* out_batch = out_edge + 2 * (size_t)Ee;   // [BK]

  float* hbuf     = (float*)d_ws;                 // N*128 (agg, then h in-place)
  float* cnt      = hbuf + (size_t)Nn * HF;       // N
  float* score    = cnt + Nn;                     // N
  float* colstats = score + Nn;                   // 256 (sum | sumsq)
  float* tscore   = colstats + 2 * HF;            // BK
  int*   node_map = (int*)(tscore + BK);          // N
  int*   perm     = node_map + Nn;                // BK
  // total ws use ~34.8 MB

  k0_init   <<<(Nn * HF + 255) / 256, 256, 0, stream>>>(hbuf, cnt, colstats, node_map);
  k1_scatter<<<Ee / 2, 256, 0, stream>>>(x, eidx, hbuf, cnt);
  k2_gemm   <<<Nn / GROWS, 256, 0, stream>>>(x, Wl, bl, Wr, cnt, hbuf);
  k3_stats  <<<Nn / 128, 256, 0, stream>>>(hbuf, colstats, colstats + HF);
  k4_bn     <<<Nn, 128, 0, stream>>>(colstats, colstats + HF, gamma, beta, attn, hbuf, score);
  k5_topk   <<<Bb, PERg, 0, stream>>>(score, node_map, perm, tscore, out_batch);
  k6_gather <<<Bb, 1024, 0, stream>>>(hbuf, perm, tscore, out_hp, out_flat);
  k7_edges  <<<(Ee + 255) / 256, 256, 0, stream>>>(eidx, node_map, out_edge);
}